// USMSharp_21311627723253
// MI455X (gfx1250) — compile-verified
//
#include <hip/hip_runtime.h>
#include <hip/hip_bf16.h>

typedef __attribute__((ext_vector_type(2))) float v2f;
typedef __attribute__((ext_vector_type(8))) float v8f;

#define W       512
#define HH      512
#define NPLANE  48          // 16*3
#define PLANE   (W*HH)
#define KS      51
#define RAD     25
#define NCHUNK  17          // K padded to 68 = 17 * 4
#define TCOLS   180         // 128 outputs + 50 taps + 2 slack, even pitch -> aligned b64 LDS reads

// CDNA5 async global->LDS path (falls back to load+store staging if builtins absent)
#if defined(__AMDGCN__) && __has_builtin(__builtin_amdgcn_global_load_async_to_lds_b32) && \
    __has_builtin(__builtin_amdgcn_s_wait_asynccnt)
#define USE_ASYNC_LDS 1
typedef __attribute__((address_space(1))) int as1_int;   // global
typedef __attribute__((address_space(3))) int as3_int;   // LDS
#else
#define USE_ASYNC_LDS 0
#endif

__device__ __forceinline__ int refl(int i) {
    i = (i < 0) ? -i : i;             // jnp 'reflect' (no edge duplication), pad=25 < 512
    return (i >= W) ? (2 * (W - 1) - i) : i;
}

// Build normalized 1-D gaussian from row 25 of the 2-D kernel: k2d = outer(g,g),
// sum(k2d[25,:]) = g[25], so g[t] = k2d[25][t] / sum(k2d[25,:]).
__device__ __forceinline__ void build_g(const float* __restrict__ k2d, float* gk, int tid) {
    if (tid < KS) {
        const float* row = k2d + 25 * KS;
        float s = 0.f;
        for (int t = 0; t < KS; ++t) s += row[t];
        gk[tid] = row[tid] / s;
    }
}

// ---------------- Horizontal blur: out[y][x] = sum_t g[t] in[y][x+t-25] ----------------
// D[m][n] = sum_k A[m][k] * B[k][n];  A = input rows (from LDS band), B = Toeplitz g[k-n].
__global__ __launch_bounds__(256) void hblur_k(const float* __restrict__ in,
                                               const float* __restrict__ k2d,
                                               float* __restrict__ out) {
    __shared__ float gk[KS];
    __shared__ float tile[16 * TCOLS];

    const int tid  = threadIdx.x;
    build_g(k2d, gk, tid);

    const int p  = blockIdx.z;
    const int y0 = blockIdx.y * 16;
    const int xb = blockIdx.x * 128;
    const float* ip = in + (size_t)p * PLANE;

#if USE_ASYNC_LDS
    // Direct DMA of the 16 x 180 input band into LDS (no VGPR round-trip, ASYNCcnt-tracked)
    for (int i = tid; i < 16 * TCOLS; i += 256) {
        int r  = i / TCOLS;
        int c  = i - r * TCOLS;
        int gc = refl(xb - RAD + c);
        __builtin_amdgcn_global_load_async_to_lds_b32(
            (as1_int*)(ip + (size_t)(y0 + r) * W + gc),
            (as3_int*)(&tile[r * TCOLS + c]), 0, 0);
    }
#else
    for (int i = tid; i < 16 * TCOLS; i += 256) {
        int r  = i / TCOLS;
        int c  = i - r * TCOLS;
        int gc = refl(xb - RAD + c);
        tile[r * TCOLS + c] = ip[(size_t)(y0 + r) * W + gc];
    }
#endif

    const int lane = tid & 31;
    const int wv   = tid >> 5;         // 8 waves per block
    const int half = lane >> 4;        // 0: lanes 0-15, 1: lanes 16-31
    const int mrow = lane & 15;

    // B Toeplitz fragments built while the band is in flight:
    // B[k][n] = g[k-n]; lane holds n=mrow, k = 4*kc + 2*half + {0,1}
    __builtin_amdgcn_s_barrier_signal(-1);   // gk ready (written before signal in program order)
    __builtin_amdgcn_s_barrier_wait(-1);
    v2f bf[NCHUNK];
    for (int kc = 0; kc < NCHUNK; ++kc) {
        int k0 = 4 * kc + 2 * half;
        int d0 = k0 - mrow, d1 = k0 + 1 - mrow;
        bf[kc].x = (d0 >= 0 && d0 < KS) ? gk[d0] : 0.f;
        bf[kc].y = (d1 >= 0 && d1 < KS) ? gk[d1] : 0.f;
    }

#if USE_ASYNC_LDS
    __builtin_amdgcn_s_wait_asynccnt(0);   // own DMA portion landed in LDS ...
#endif
    __syncthreads();                       // ... then all waves' portions visible

    // A fragments from LDS: row m=mrow, cols wv*16 + k
    v8f acc = {};
    const int abase = mrow * TCOLS + wv * 16 + 2 * half;   // even -> 8B aligned
    for (int kc = 0; kc < NCHUNK; ++kc) {
        v2f a = *(const v2f*)(&tile[abase + 4 * kc]);
        acc = __builtin_amdgcn_wmma_f32_16x16x4_f32(false, a, false, bf[kc],
                                                    (short)0, acc, false, false);
    }

    // D: VGPR r holds row y0+r (lanes 0-15) / y0+r+8 (lanes 16-31), cols contiguous
    float* op = out + (size_t)p * PLANE;
    const int xo = xb + wv * 16 + mrow;
    for (int r = 0; r < 8; ++r)
        op[(size_t)(y0 + r + half * 8) * W + xo] = acc[r];
}

// ---------------- Vertical blur (+ fused elementwise epilogues) ----------------
// D[m][n] = sum_k A[m][k] * B[k][n];  A = Toeplitz g[k-m] (regs), B = input rows from global.
// mode 0: blur = vblur(in); res = x - blur; write mask -> out0, clip(x+0.5*res) -> out1
// mode 1: soft = vblur(in); out1 = soft*out1 + (1-soft)*x   (out1 read+write = d_out)
__global__ __launch_bounds__(256) void vblur_k(const float* __restrict__ in,
                                               const float* __restrict__ k2d,
                                               const float* __restrict__ x,
                                               float* __restrict__ out0,
                                               float* __restrict__ out1,
                                               int mode) {
    __shared__ float gk[KS];
    const int tid = threadIdx.x;
    build_g(k2d, gk, tid);
    __syncthreads();

    const int lane = tid & 31;
    const int wv   = tid >> 5;
    const int half = lane >> 4;
    const int mrow = lane & 15;

    // A Toeplitz fragments: A[m][k] = g[k-m]; lane holds m=mrow, k = 4*kc + 2*half + {0,1}
    v2f af[NCHUNK];
    for (int kc = 0; kc < NCHUNK; ++kc) {
        int k0 = 4 * kc + 2 * half;
        int d0 = k0 - mrow, d1 = k0 + 1 - mrow;
        af[kc].x = (d0 >= 0 && d0 < KS) ? gk[d0] : 0.f;
        af[kc].y = (d1 >= 0 && d1 < KS) ? gk[d1] : 0.f;
    }

    const int p  = blockIdx.z;
    const int y0 = blockIdx.y * 16;
    const int x0 = blockIdx.x * 128 + wv * 16;
    const float* ip = in + (size_t)p * PLANE;

    const size_t pbase = (size_t)p * PLANE;
    const int col = x0 + mrow;

    // Prefetch epilogue operands (x rows, and d_out rows in blend mode) so their
    // latency overlaps the 17-WMMA chain. global_prefetch_b8: whole line per lane.
    for (int r = 0; r < 8; ++r) {
        const size_t idx = pbase + (size_t)(y0 + r + half * 8) * W + col;
        __builtin_prefetch(&x[idx], 0, 3);
        if (mode != 0) __builtin_prefetch(&out1[idx], 0, 3);
    }

    // B fragments straight from global: row k, cols x0..x0+15 (64B contiguous per half-wave)
    v8f acc = {};
    for (int kc = 0; kc < NCHUNK; ++kc) {
        int k0 = 4 * kc + 2 * half;
        v2f b;
        b.x = ip[(size_t)refl(y0 - RAD + k0)     * W + col];
        b.y = ip[(size_t)refl(y0 - RAD + k0 + 1) * W + col];
        acc = __builtin_amdgcn_wmma_f32_16x16x4_f32(false, af[kc], false, b,
                                                    (short)0, acc, false, false);
    }

    for (int r = 0; r < 8; ++r) {
        const size_t idx = pbase + (size_t)(y0 + r + half * 8) * W + col;
        const float xv = x[idx];
        const float bl = acc[r];
        if (mode == 0) {
            float res  = xv - bl;
            float mask = (fabsf(res) * 255.0f > 10.0f) ? 1.0f : 0.0f;
            float oc   = fminf(fmaxf(xv + 0.5f * res, 0.0f), 1.0f);
            out0[idx] = mask;   // mask for second blur
            out1[idx] = oc;     // clipped sharpened image -> d_out
        } else {
            float oc = out1[idx];
            out1[idx] = bl * oc + (1.0f - bl) * xv;
        }
    }
}

extern "C" void kernel_launch(void* const* d_in, const int* in_sizes, int n_in,
                              void* d_out, int out_size, void* d_ws, size_t ws_size,
                              hipStream_t stream) {
    (void)in_sizes; (void)n_in; (void)out_size; (void)ws_size;
    const float* x   = (const float*)d_in[0];   // [16,3,512,512] f32
    const float* k2d = (const float*)d_in[1];   // [51,51] f32
    float* out = (float*)d_out;

    const size_t nelem = (size_t)NPLANE * PLANE;   // 12,582,912
    float* ws0 = (float*)d_ws;                     // h-blur temp (reused)
    float* ws1 = ws0 + nelem;                      // mask

    dim3 grid(4, 32, NPLANE);   // 4 * 128 cols, 32 * 16 rows, 48 planes
    dim3 block(256);            // 8 wave32

    // usm_blur = vblur(hblur(x)); fused: residual, mask, clip
    hblur_k<<<grid, block, 0, stream>>>(x, k2d, ws0);
    vblur_k<<<grid, block, 0, stream>>>(ws0, k2d, x, ws1, out, 0);
    // soft_mask = vblur(hblur(mask)); fused final blend into d_out
    hblur_k<<<grid, block, 0, stream>>>(ws1, k2d, ws0);
    vblur_k<<<grid, block, 0, stream>>>(ws0, k2d, x, ws1, out, 1);
}